// NCC_mok_730144441165
// MI455X (gfx1250) — compile-verified
//
#include <hip/hip_runtime.h>

typedef __attribute__((ext_vector_type(2))) float v2f;
typedef __attribute__((ext_vector_type(8))) float v8f;

#define WIN     9
#define PAD     4
#define TD      16          // interior tile edge
#define TIN     24          // TD + 2*PAD
#define DVOL    192
#define HVOL    224
#define WVOL    192
#define NVOX    (192*224*192)

// LDS layout (floats)
#define SI_OFF  0
#define SJ_OFF  (SI_OFF + TIN*TIN*TIN)          // 13824
#define S1_OFF  (SJ_OFF + TIN*TIN*TIN)          // +13824  S1: [line(d*24+h)][wo]  576*16
#define S2_OFF  (S1_OFF + TIN*TIN*TD)           // +9216   S2: [line(d*16+w)][ho]  384*16
#define BS_OFF  (S2_OFF + TIN*TD*TD)            // +6144   block sum
#define SMEM_FLOATS (BS_OFF + 32)

__device__ __forceinline__ float band(int p, int m) {
    // ones band matrix: out[m] = sum_{k=0..8} in[m+k]  => in position p feeds out m iff m <= p <= m+8
    return (p >= m && p <= m + 8) ? 1.0f : 0.0f;
}

__global__ void __launch_bounds__(256) ncc_zero_out(float* out) {
    if (threadIdx.x == 0) out[0] = 0.0f;
}

__global__ void __launch_bounds__(256) ncc_fused_kernel(const float* __restrict__ I,
                                                        const float* __restrict__ J,
                                                        float* __restrict__ out) {
    extern __shared__ float smem[];
    float* sI = smem + SI_OFF;
    float* sJ = smem + SJ_OFF;
    float* S1 = smem + S1_OFF;
    float* S2 = smem + S2_OFF;
    float* bsum = smem + BS_OFF;

    const int tid  = threadIdx.x;
    const int lane = tid & 31;
    const int wave = tid >> 5;
    const int n16  = lane & 15;        // N (and M for A-operand) coordinate owned by this lane
    const int half = lane >> 4;        // 0: lanes 0-15, 1: lanes 16-31
    const int kb   = half * 2;         // K base for A/B fragments (K=0,1 vs K=2,3)
    const int hi8  = half * 8;         // M offset for C/D fragments

    const int d0 = (int)blockIdx.z * TD - PAD;
    const int h0 = (int)blockIdx.y * TD - PAD;
    const int w0 = (int)blockIdx.x * TD - PAD;

    if (tid == 0) *bsum = 0.0f;

    // ---- load 24^3 halo tiles of I and J (zero-pad out-of-volume) ----
    for (int idx = tid; idx < TIN * TIN * TIN; idx += 256) {
        int ld = idx / (TIN * TIN);
        int lh = (idx / TIN) % TIN;
        int lw = idx % TIN;
        int gd = d0 + ld, gh = h0 + lh, gw = w0 + lw;
        bool inr = (gd >= 0 && gd < DVOL && gh >= 0 && gh < HVOL && gw >= 0 && gw < WVOL);
        long g = ((long)gd * HVOL + gh) * (long)WVOL + gw;
        sI[idx] = inr ? I[g] : 0.0f;
        sJ[idx] = inr ? J[g] : 0.0f;
    }
    __syncthreads();

    // per-lane final box-sums for the 5 fields, 2 D-stage tiles, 8 C/D regs each
    float res[2][5][8];

    #pragma unroll
    for (int f = 0; f < 5; ++f) {
        // ================= stage W: filter along W =================
        // lines = (d,h) in [0,24)x[0,24) -> 576 lines, 36 N-tiles; M = wo in [0,16)
        for (int t = wave; t < 36; t += 8) {
            v8f acc = {0.f,0.f,0.f,0.f,0.f,0.f,0.f,0.f};
            const int line = t * 16 + n16;
            #pragma unroll
            for (int c = 0; c < 6; ++c) {
                const int p0 = 4 * c + kb;
                v2f a; a.x = band(p0, n16); a.y = band(p0 + 1, n16);
                const int base = line * TIN + p0;
                float i0 = sI[base], i1 = sI[base + 1];
                float j0 = sJ[base], j1 = sJ[base + 1];
                v2f b;
                if      (f == 0) { b.x = i0;      b.y = i1;      }
                else if (f == 1) { b.x = j0;      b.y = j1;      }
                else if (f == 2) { b.x = i0 * i0; b.y = i1 * i1; }
                else if (f == 3) { b.x = j0 * j0; b.y = j1 * j1; }
                else             { b.x = i0 * j0; b.y = i1 * j1; }
                acc = __builtin_amdgcn_wmma_f32_16x16x4_f32(false, a, false, b,
                                                            (short)0, acc, false, false);
            }
            #pragma unroll
            for (int r = 0; r < 8; ++r)
                S1[line * TD + (r + hi8)] = acc[r];
        }
        __syncthreads();

        // ================= stage H: filter along H =================
        // lines = (d,w) in [0,24)x[0,16) -> 384 lines, 24 N-tiles; M = ho in [0,16)
        for (int t = wave; t < 24; t += 8) {
            v8f acc = {0.f,0.f,0.f,0.f,0.f,0.f,0.f,0.f};
            const int lineId = t * 16 + n16;
            const int dd = lineId >> 4;
            const int ww = lineId & 15;
            #pragma unroll
            for (int c = 0; c < 6; ++c) {
                const int p0 = 4 * c + kb;
                v2f a; a.x = band(p0, n16); a.y = band(p0 + 1, n16);
                v2f b;
                b.x = S1[(dd * TIN + p0) * TD + ww];
                b.y = S1[(dd * TIN + p0 + 1) * TD + ww];
                acc = __builtin_amdgcn_wmma_f32_16x16x4_f32(false, a, false, b,
                                                            (short)0, acc, false, false);
            }
            #pragma unroll
            for (int r = 0; r < 8; ++r)
                S2[lineId * TD + (r + hi8)] = acc[r];
        }
        __syncthreads();

        // ================= stage D: filter along D =================
        // lines = (h,w) in [0,16)x[0,16) -> 256 lines, 16 N-tiles; M = do in [0,16)
        #pragma unroll
        for (int s = 0; s < 2; ++s) {
            const int t = wave + 8 * s;
            v8f acc = {0.f,0.f,0.f,0.f,0.f,0.f,0.f,0.f};
            const int lineId = t * 16 + n16;
            const int hh = lineId >> 4;
            const int ww = lineId & 15;
            #pragma unroll
            for (int c = 0; c < 6; ++c) {
                const int p0 = 4 * c + kb;
                v2f a; a.x = band(p0, n16); a.y = band(p0 + 1, n16);
                v2f b;
                b.x = S2[(p0 * TD + ww) * TD + hh];
                b.y = S2[((p0 + 1) * TD + ww) * TD + hh];
                acc = __builtin_amdgcn_wmma_f32_16x16x4_f32(false, a, false, b,
                                                            (short)0, acc, false, false);
            }
            #pragma unroll
            for (int r = 0; r < 8; ++r)
                res[s][f][r] = acc[r];
        }
        __syncthreads();   // S1/S2 reused by next field
    }

    // ================= elementwise NCC combine (layout-agnostic: all fields share C/D layout) =========
    const float winsz = 729.0f;     // 9^3
    float local = 0.0f;
    #pragma unroll
    for (int s = 0; s < 2; ++s) {
        #pragma unroll
        for (int r = 0; r < 8; ++r) {
            float Is = res[s][0][r], Js = res[s][1][r];
            float I2 = res[s][2][r], J2 = res[s][3][r], IJ = res[s][4][r];
            float uI = Is * (1.0f / winsz);
            float uJ = Js * (1.0f / winsz);
            float cross = IJ - uJ * Is - uI * Js + uI * uJ * winsz;
            float Iv = I2 - 2.0f * uI * Is + uI * uI * winsz;
            float Jv = J2 - 2.0f * uJ * Js + uJ * uJ * winsz;
            float cc = cross * cross / (Iv * Jv + 1e-5f);
            local += cc;
        }
    }

    atomicAdd(bsum, local);                 // LDS ds_add_f32
    __syncthreads();
    if (tid == 0)
        atomicAdd(out, (*bsum) * (-1.0f / (float)NVOX));
}

extern "C" void kernel_launch(void* const* d_in, const int* in_sizes, int n_in,
                              void* d_out, int out_size, void* d_ws, size_t ws_size,
                              hipStream_t stream) {
    const float* I = (const float*)d_in[0];
    const float* J = (const float*)d_in[1];
    float* out = (float*)d_out;

    ncc_zero_out<<<1, 256, 0, stream>>>(out);

    dim3 grid(WVOL / TD, HVOL / TD, DVOL / TD);   // 12 x 14 x 12
    size_t shmem = (size_t)SMEM_FLOATS * sizeof(float);   // ~172 KB of the 320 KB WGP LDS
    ncc_fused_kernel<<<grid, 256, shmem, stream>>>(I, J, out);
}